// GraphInteractionLayer_4810363372965
// MI455X (gfx1250) — compile-verified
//
#include <hip/hip_runtime.h>
#include <hip/hip_bf16.h>
#include <math.h>

// ---------------------------------------------------------------------------
// CDNA5 / gfx1250, wave32. FP32 WMMA path: V_WMMA_F32_16X16X4_F32.
//   A (16x4, M x K): lane 0-15  -> M=lane,    v0=K0, v1=K1
//                    lane 16-31 -> M=lane-16, v0=K2, v1=K3
//   B (4x16, K x N): mirrored: lane half selects K pair {0,1}/{2,3}, N=lane&15
//   C/D (16x16):     component r -> row r + 8*(lane>>4), col = lane&15
// Large GEMMs: Tensor Data Mover stages A/B tiles into LDS (double-buffered),
// tracked with TENSORcnt; inner loop feeds WMMA entirely from LDS.
// ---------------------------------------------------------------------------

typedef float v2f __attribute__((ext_vector_type(2)));
typedef float v8f __attribute__((ext_vector_type(8)));
typedef unsigned int v4u __attribute__((ext_vector_type(4)));
typedef int v4i __attribute__((ext_vector_type(4)));
typedef int v8i __attribute__((ext_vector_type(8)));

#define BATCH 32
#define NN    128
#define DD    512
#define EPSLN 1e-5f

// macro-tile geometry for the TDM GEMM
#define MT 128
#define NTILE 64
#define KT 32

__device__ __forceinline__ v8f wmma_f32(v2f a, v2f b, v8f c) {
  return __builtin_amdgcn_wmma_f32_16x16x4_f32(
      /*neg_a=*/false, a, /*neg_b=*/false, b,
      /*c_mod=*/(short)0, c, /*reuse_a=*/false, /*reuse_b=*/false);
}

// ---------------------------------------------------------------------------
// Issue a TDM 2D tile load: global (tile_h rows x tile_w elems, row stride
// stride_elems, fp32) -> LDS at lds_byte_off, packed row-major (pitch tile_w).
// D# built per CDNA5 ISA 8.3/8.4: group0 {count=1, lds_addr, global_addr,
// type=2}; group1 {data_size=4B, tensor dims == tile dims, real row stride}.
// ---------------------------------------------------------------------------
__device__ __forceinline__ void tdm_load_tile_2d(unsigned lds_byte_off,
                                                 const float* gptr,
                                                 unsigned tile_w, unsigned tile_h,
                                                 unsigned stride_elems)
{
  unsigned long long ga = (unsigned long long)(uintptr_t)gptr;
  v4u g0;
  g0.x = 1u;                                             // count=1 (user mode)
  g0.y = lds_byte_off;                                   // lds_addr (bytes)
  g0.z = (unsigned)ga;                                   // global_addr[31:0]
  g0.w = (unsigned)((ga >> 32) & 0x01FFFFFFull)          // global_addr[56:32]
         | (2u << 30);                                   // type=2 ("image")
  const unsigned long long s0 = (unsigned long long)stride_elems;
  v8i g1;
  g1[0] = (int)(2u << 16);                               // data_size=2 -> 4 bytes
  g1[1] = (int)((tile_w & 0xFFFFu) << 16);               // tensor_dim0[15:0]
  g1[2] = (int)((tile_w >> 16) | ((tile_h & 0xFFFFu) << 16)); // dim0 hi | dim1 lo
  g1[3] = (int)((tile_h >> 16) | (tile_w << 16));        // dim1 hi | tile_dim0
  g1[4] = (int)(tile_h & 0xFFFFu);                       // tile_dim1; tile_dim2=0
  g1[5] = (int)(unsigned)(s0 & 0xFFFFFFFFull);           // dim0_stride[31:0]
  g1[6] = (int)((unsigned)(s0 >> 32) & 0xFFFFu);         // dim0_stride[47:32]
  g1[7] = 0;
  v4i z4 = {0, 0, 0, 0};
#if defined(__clang_major__) && (__clang_major__ >= 23)
  v8i z8 = {0, 0, 0, 0, 0, 0, 0, 0};
  __builtin_amdgcn_tensor_load_to_lds(g0, g1, z4, z4, z8, 0);
#else
  __builtin_amdgcn_tensor_load_to_lds(g0, g1, z4, z4, 0);
#endif
}

// ---------------------------------------------------------------------------
// TDM-fed, double-buffered GEMM: C[M,N] = act(A[M,K] @ B[K,N] + bias)
// Block: 256 threads (8 waves) -> 128x64 macro-tile; wave w owns a 16x64 strip.
// Requires M%128==0, N%64==0, K%32==0.
// ---------------------------------------------------------------------------
__global__ __launch_bounds__(256)
void gemm_tdm_wmma(const float* __restrict__ A, const float* __restrict__ B,
                   const float* __restrict__ bias, float* __restrict__ C,
                   int M, int N, int K, int relu)
{
  __shared__ float smem[2 * (MT * KT + KT * NTILE)];     // 2 x 6144 floats = 48 KB
  const unsigned ldsbase = (unsigned)(uintptr_t)(&smem[0]); // flat[31:0] = LDS offset

  const int lane = threadIdx.x & 31;
  const int wave = threadIdx.x >> 5;                     // 0..7
  const int half = lane >> 4, l = lane & 15;

  const int colBase = blockIdx.x * NTILE;
  const int rowBase = blockIdx.y * MT;
  const int nsteps  = K / KT;
  const unsigned bufElems = MT * KT + KT * NTILE;        // 6144
  const unsigned bOff     = MT * KT;                     // 4096 (B tile offset)

  // prologue: stage K-step 0 into buffer 0
  if (wave == 0) {
    tdm_load_tile_2d(ldsbase,            A + (size_t)rowBase * K, KT, MT, (unsigned)K);
    tdm_load_tile_2d(ldsbase + bOff * 4, B + (size_t)colBase,     NTILE, KT, (unsigned)N);
  }

  v8f acc[4];
  #pragma unroll
  for (int t = 0; t < 4; ++t)
    #pragma unroll
    for (int r = 0; r < 8; ++r) acc[t][r] = 0.0f;

  for (int i = 0; i < nsteps; ++i) {
    const int cur = i & 1;
    if (wave == 0) {
      if (i + 1 < nsteps) {
        const int kk = (i + 1) * KT;
        const unsigned nb = (unsigned)(cur ^ 1) * bufElems * 4u;
        tdm_load_tile_2d(ldsbase + nb,            A + (size_t)rowBase * K + kk, KT, MT, (unsigned)K);
        tdm_load_tile_2d(ldsbase + nb + bOff * 4, B + (size_t)kk * N + colBase, NTILE, KT, (unsigned)N);
        __builtin_amdgcn_s_wait_tensorcnt((short)2);     // current pair complete
      } else {
        __builtin_amdgcn_s_wait_tensorcnt((short)0);     // drain
      }
    }
    __syncthreads();                                     // tile visible to all waves

    const float* __restrict__ At = &smem[cur * bufElems];
    const float* __restrict__ Bt = &smem[cur * bufElems + bOff];
    const float* __restrict__ arow = At + (size_t)(wave * 16 + l) * KT;

    #pragma unroll
    for (int k4 = 0; k4 < KT; k4 += 4) {
      const int k0 = k4 + half * 2;
      v2f a; a.x = arow[k0]; a.y = arow[k0 + 1];
      #pragma unroll
      for (int t = 0; t < 4; ++t) {
        v2f b;
        b.x = Bt[k0 * NTILE + t * 16 + l];
        b.y = Bt[(k0 + 1) * NTILE + t * 16 + l];
        acc[t] = wmma_f32(a, b, acc[t]);
      }
    }
    __syncthreads();                                     // safe to overwrite buffer
  }

  const int row0 = rowBase + wave * 16;
  #pragma unroll
  for (int t = 0; t < 4; ++t) {
    const int col = colBase + t * 16 + l;
    const float bv = bias ? bias[col] : 0.0f;
    #pragma unroll
    for (int r = 0; r < 8; ++r) {
      float val = acc[t][r] + bv;
      if (relu) val = fmaxf(val, 0.0f);
      C[(size_t)(row0 + r + 8 * half) * N + col] = val;
    }
  }
}

// ---------------------------------------------------------------------------
// Small GEMM (gamma/beta, M=32): direct-from-global WMMA, one tile per wave.
// ---------------------------------------------------------------------------
__global__ __launch_bounds__(128)
void gemm_nn_wmma(const float* __restrict__ A, const float* __restrict__ B,
                  const float* __restrict__ bias, float* __restrict__ C,
                  int M, int N, int K, int relu)
{
  const int lane  = threadIdx.x & 31;
  const int wave  = threadIdx.x >> 5;
  const int tilesN = N >> 4;
  const int tile = blockIdx.x * 4 + wave;
  const int tm = tile / tilesN;
  const int tn = tile - tm * tilesN;
  if (tm * 16 >= M) return;                 // wave-uniform: EXEC stays all-1

  const int half = lane >> 4;
  const int l    = lane & 15;
  const int rowA = tm * 16 + l;
  const int colB = tn * 16 + l;

  v8f c;
  #pragma unroll
  for (int r = 0; r < 8; ++r) c[r] = 0.0f;

  const float* __restrict__ Arow = A + (size_t)rowA * K;
  #pragma unroll 8
  for (int kk = 0; kk < K; kk += 4) {
    const int k0 = kk + half * 2;
    v2f a; a.x = Arow[k0];                 a.y = Arow[k0 + 1];
    v2f b; b.x = B[(size_t)k0 * N + colB]; b.y = B[(size_t)(k0 + 1) * N + colB];
    c = wmma_f32(a, b, c);
  }

  const float bv = bias ? bias[colB] : 0.0f;
  #pragma unroll
  for (int r = 0; r < 8; ++r) {
    float val = c[r] + bv;
    if (relu) val = fmaxf(val, 0.0f);
    C[(size_t)(tm * 16 + r + 8 * half) * N + colB] = val;
  }
}

// ---------------------------------------------------------------------------
// Per-batch logits: L[b] = (Q[b] @ K[b]^T) * scale + biasNN[b]   (M=N=128,K=512)
// ---------------------------------------------------------------------------
__global__ __launch_bounds__(128)
void qk_wmma(const float* __restrict__ Q, const float* __restrict__ Km,
             const float* __restrict__ biasNN, float* __restrict__ L, float scale)
{
  const int b    = blockIdx.y;
  const int lane = threadIdx.x & 31;
  const int wave = threadIdx.x >> 5;
  const int tile = blockIdx.x * 4 + wave;   // 64 tiles (8x8)
  const int tm = tile >> 3, tn = tile & 7;
  const int half = lane >> 4, l = lane & 15;

  const float* __restrict__ Qb = Q + (size_t)b * NN * DD;
  const float* __restrict__ Kb = Km + (size_t)b * NN * DD;

  const float* __restrict__ qrow = Qb + (size_t)(tm * 16 + l) * DD;
  const float* __restrict__ krow = Kb + (size_t)(tn * 16 + l) * DD;  // B[k][n]=K[n][k]

  v8f c;
  #pragma unroll
  for (int r = 0; r < 8; ++r) c[r] = 0.0f;

  #pragma unroll 8
  for (int kk = 0; kk < DD; kk += 4) {
    const int k0 = kk + half * 2;
    v2f a; a.x = qrow[k0]; a.y = qrow[k0 + 1];
    v2f bb; bb.x = krow[k0]; bb.y = krow[k0 + 1];
    c = wmma_f32(a, bb, c);
  }

  float* __restrict__ Lb = L + (size_t)b * NN * NN;
  const float* __restrict__ Bb = biasNN + (size_t)b * NN * NN;
  #pragma unroll
  for (int r = 0; r < 8; ++r) {
    const int row = tm * 16 + r + 8 * half;
    const int col = tn * 16 + l;
    Lb[row * NN + col] = c[r] * scale + Bb[row * NN + col];
  }
}

// ---------------------------------------------------------------------------
// Per-batch aggregation: agg[b] = (W[b] @ V[b]) * mask[b,row]   (M=128,K=128,N=512)
// ---------------------------------------------------------------------------
__global__ __launch_bounds__(128)
void agg_wmma(const float* __restrict__ W, const float* __restrict__ V,
              const float* __restrict__ pmask, float* __restrict__ AGG)
{
  const int b    = blockIdx.y;
  const int lane = threadIdx.x & 31;
  const int wave = threadIdx.x >> 5;
  const int tile = blockIdx.x * 4 + wave;   // 256 tiles (8 x 32)
  const int tilesN = DD >> 4;               // 32
  const int tm = tile / tilesN, tn = tile - tm * tilesN;
  const int half = lane >> 4, l = lane & 15;

  const float* __restrict__ Wb = W + (size_t)b * NN * NN;
  const float* __restrict__ Vb = V + (size_t)b * NN * DD;

  const float* __restrict__ wrow = Wb + (size_t)(tm * 16 + l) * NN;
  const int colB = tn * 16 + l;

  v8f c;
  #pragma unroll
  for (int r = 0; r < 8; ++r) c[r] = 0.0f;

  #pragma unroll 8
  for (int kk = 0; kk < NN; kk += 4) {
    const int k0 = kk + half * 2;
    v2f a; a.x = wrow[k0]; a.y = wrow[k0 + 1];
    v2f bb; bb.x = Vb[(size_t)k0 * DD + colB]; bb.y = Vb[(size_t)(k0 + 1) * DD + colB];
    c = wmma_f32(a, bb, c);
  }

  float* __restrict__ Ab = AGG + (size_t)b * NN * DD;
  #pragma unroll
  for (int r = 0; r < 8; ++r) {
    const int row = tm * 16 + r + 8 * half;
    Ab[(size_t)row * DD + colB] = c[r] * pmask[b * NN + row];
  }
}

// ---------------------------------------------------------------------------
// Block-of-256 sum reduction (LDS)
// ---------------------------------------------------------------------------
__device__ __forceinline__ float block_sum256(float v, float* red) {
  const int t = threadIdx.x;
  red[t] = v; __syncthreads();
  #pragma unroll
  for (int s = 128; s > 0; s >>= 1) {
    if (t < s) red[t] += red[t + s];
    __syncthreads();
  }
  const float r = red[0]; __syncthreads();
  return r;
}

// nf0 = LN(x; cn_g,cn_b) * (1+gamma[b]) + beta[b]     one block per (b,n) row
__global__ __launch_bounds__(256)
void film_ln(const float* __restrict__ X, const float* __restrict__ g,
             const float* __restrict__ bta, const float* __restrict__ gamma,
             const float* __restrict__ beta, float* __restrict__ Y)
{
  __shared__ float red[256];
  const int row = blockIdx.x;              // b*NN + n
  const int b = row >> 7;
  const int t = threadIdx.x;
  const float* __restrict__ x = X + (size_t)row * DD;
  const float x0 = x[t], x1 = x[t + 256];

  const float mean = block_sum256(x0 + x1, red) * (1.0f / DD);
  const float d0 = x0 - mean, d1 = x1 - mean;
  const float var = block_sum256(d0 * d0 + d1 * d1, red) * (1.0f / DD);
  const float inv = rsqrtf(var + EPSLN);

  float* __restrict__ y = Y + (size_t)row * DD;
  const float* __restrict__ gm = gamma + (size_t)b * DD;
  const float* __restrict__ bt = beta  + (size_t)b * DD;
  {
    const float ln = d0 * inv * g[t] + bta[t];
    y[t] = ln * (1.0f + gm[t]) + bt[t];
  }
  {
    const int j = t + 256;
    const float ln = d1 * inv * g[j] + bta[j];
    y[j] = ln * (1.0f + gm[j]) + bt[j];
  }
}

// Y = LN(A + R; g,b) [ * mask[row] ]       one block per row; Y may alias A
__global__ __launch_bounds__(256)
void residual_ln(const float* __restrict__ A, const float* __restrict__ R,
                 const float* __restrict__ g, const float* __restrict__ bb,
                 const float* __restrict__ pmask, float* __restrict__ Y)
{
  __shared__ float red[256];
  const int row = blockIdx.x;
  const int t = threadIdx.x;
  const float x0 = A[(size_t)row * DD + t]       + R[(size_t)row * DD + t];
  const float x1 = A[(size_t)row * DD + t + 256] + R[(size_t)row * DD + t + 256];

  const float mean = block_sum256(x0 + x1, red) * (1.0f / DD);
  const float d0 = x0 - mean, d1 = x1 - mean;
  const float var = block_sum256(d0 * d0 + d1 * d1, red) * (1.0f / DD);
  const float inv = rsqrtf(var + EPSLN);
  const float mk = pmask ? pmask[row] : 1.0f;

  Y[(size_t)row * DD + t]       = (d0 * inv * g[t]       + bb[t])       * mk;
  Y[(size_t)row * DD + t + 256] = (d1 * inv * g[t + 256] + bb[t + 256]) * mk;
}

// ---------------------------------------------------------------------------
// biasNN[b,n,m] = tanh(sum_d relu(dx*rW1[0,d]+dy*rW1[1,d]+rb1[d]) * rW2[d] + rb2)*2
//                 + spk_emb[edge_type]
// One block (128 threads = m) per (b,n) row; MLP weights staged in LDS.
// ---------------------------------------------------------------------------
__global__ __launch_bounds__(128)
void rel_spk_bias(const float* __restrict__ centers, const int* __restrict__ spk,
                  const float* __restrict__ rW1, const float* __restrict__ rb1,
                  const float* __restrict__ rW2, const float* __restrict__ rb2,
                  const float* __restrict__ spk_emb, float* __restrict__ biasNN)
{
  __shared__ float w0[DD], w1[DD], w2[DD], bh[DD];
  for (int i = threadIdx.x; i < DD; i += 128) {
    w0[i] = rW1[i];
    w1[i] = rW1[DD + i];
    w2[i] = rW2[i];
    bh[i] = rb1[i];
  }
  __syncthreads();

  const int row = blockIdx.x;              // b*NN + n
  const int b = row >> 7, n = row & 127;
  const int m = threadIdx.x;

  const float nx = centers[(size_t)(b * NN + n) * 2 + 0];
  const float ny = centers[(size_t)(b * NN + n) * 2 + 1];
  const float mx = centers[(size_t)(b * NN + m) * 2 + 0];
  const float my = centers[(size_t)(b * NN + m) * 2 + 1];
  const float dx = nx - mx, dy = ny - my;

  float acc = 0.0f;
  #pragma unroll 8
  for (int d = 0; d < DD; ++d) {
    float h = fmaf(dx, w0[d], fmaf(dy, w1[d], bh[d]));
    h = fmaxf(h, 0.0f);
    acc = fmaf(h, w2[d], acc);
  }
  const float rbv = tanhf(acc + rb2[0]) * 2.0f;

  const int s = spk[b];
  int et = 0;
  if (n == s) et = 1;
  if (m == s) et = 2;                      // target overrides source
  biasNN[(size_t)row * NN + m] = rbv + spk_emb[et];
}

// ---------------------------------------------------------------------------
// Masked softmax over m, in-place on logits. One block (128 threads) per (b,n).
// ---------------------------------------------------------------------------
__global__ __launch_bounds__(128)
void softmax_mask(float* __restrict__ L, const float* __restrict__ pmask)
{
  __shared__ float red[128];
  const int row = blockIdx.x;
  const int b = row >> 7, n = row & 127;
  const int m = threadIdx.x;
  float* __restrict__ l = L + (size_t)row * NN;

  const float valid = pmask[b * NN + m];
  float t = l[m];
  if (m == n)        t = -10000.0f;
  if (valid == 0.0f) t = -10000.0f;

  red[m] = t; __syncthreads();
  #pragma unroll
  for (int s = 64; s > 0; s >>= 1) { if (m < s) red[m] = fmaxf(red[m], red[m + s]); __syncthreads(); }
  const float mx = red[0]; __syncthreads();

  const float e = expf(t - mx);
  red[m] = e; __syncthreads();
  #pragma unroll
  for (int s = 64; s > 0; s >>= 1) { if (m < s) red[m] += red[m + s]; __syncthreads(); }
  const float sum = red[0];

  float w = e / sum;
  if (!(w == w) || fabsf(w) > 3.0e38f) w = 0.0f;   // nan_to_num
  if (valid == 0.0f) w = 0.0f;
  l[m] = w;
}

// ---------------------------------------------------------------------------
// Host-side orchestration
// ---------------------------------------------------------------------------
extern "C" void kernel_launch(void* const* d_in, const int* in_sizes, int n_in,
                              void* d_out, int out_size, void* d_ws, size_t ws_size,
                              hipStream_t stream) {
  (void)in_sizes; (void)n_in; (void)out_size; (void)ws_size;
  const float* node    = (const float*)d_in[0];
  const float* centers = (const float*)d_in[1];
  const float* pmask   = (const float*)d_in[2];
  const int*   spk     = (const int*)  d_in[3];
  const float* cond    = (const float*)d_in[4];
  const float* Wq  = (const float*)d_in[5];  const float* bq  = (const float*)d_in[6];
  const float* Wk  = (const float*)d_in[7];  const float* bk  = (const float*)d_in[8];
  const float* Wv  = (const float*)d_in[9];  const float* bv  = (const float*)d_in[10];
  const float* rW1 = (const float*)d_in[11]; const float* rb1 = (const float*)d_in[12];
  const float* rW2 = (const float*)d_in[13]; const float* rb2 = (const float*)d_in[14];
  const float* Wo  = (const float*)d_in[15]; const float* bo  = (const float*)d_in[16];
  const float* n1g = (const float*)d_in[17]; const float* n1b = (const float*)d_in[18];
  const float* cng = (const float*)d_in[19]; const float* cnb = (const float*)d_in[20];
  const float* Wcg = (const float*)d_in[21]; const float* bcg = (const float*)d_in[22];
  const float* Wcb = (const float*)d_in[23]; const float* bcb = (const float*)d_in[24];
  const float* fW1 = (const float*)d_in[25]; const float* fb1 = (const float*)d_in[26];
  const float* fW2 = (const float*)d_in[27]; const float* fb2 = (const float*)d_in[28];
  const float* n2g = (const float*)d_in[29]; const float* n2b = (const float*)d_in[30];
  const float* semb= (const float*)d_in[31];

  const size_t BD   = (size_t)BATCH * DD;          // 16384
  const size_t BND  = (size_t)BATCH * NN * DD;     // 2,097,152
  const size_t BNN  = (size_t)BATCH * NN * NN;     // 524,288

  float* ws     = (float*)d_ws;
  float* gamma  = ws;            ws += BD;
  float* beta   = ws;            ws += BD;
  float* nf0    = ws;            ws += BND;        // FiLM output, later nf1 (in place)
  float* qb     = ws;            ws += BND;        // q -> agg -> ff
  float* kb     = ws;            ws += BND;        // k -> o (agg@Wo)
  float* vb     = ws;            ws += BND;        // v
  float* biasNN = ws;            ws += BNN;        // rel + speaker bias
  float* logit  = ws;            ws += BNN;        // logits -> attention weights
  float* hb     = ws;            ws += 2 * BND;    // FFN hidden [4096,1024]

  const int ROWS = BATCH * NN;   // 4096

  auto gemm_small = [&](const float* A, const float* Bm, const float* bias, float* C,
                        int M, int N, int K, int relu) {
    const int tiles = (M / 16) * (N / 16);
    gemm_nn_wmma<<<(tiles + 3) / 4, 128, 0, stream>>>(A, Bm, bias, C, M, N, K, relu);
  };
  auto gemm_big = [&](const float* A, const float* Bm, const float* bias, float* C,
                      int M, int N, int K, int relu) {
    gemm_tdm_wmma<<<dim3(N / NTILE, M / MT), 256, 0, stream>>>(A, Bm, bias, C, M, N, K, relu);
  };

  // 1) FiLM params: gamma/beta = cond @ Wcg/Wcb + b    (M=32: small path)
  gemm_small(cond, Wcg, bcg, gamma, BATCH, DD, DD, 0);
  gemm_small(cond, Wcb, bcb, beta,  BATCH, DD, DD, 0);

  // 2) nf0 = LN(x)*(1+gamma)+beta
  film_ln<<<ROWS, 256, 0, stream>>>(node, cng, cnb, gamma, beta, nf0);

  // 3) QKV projections (TDM-fed)
  gemm_big(nf0, Wq, bq, qb, ROWS, DD, DD, 0);
  gemm_big(nf0, Wk, bk, kb, ROWS, DD, DD, 0);
  gemm_big(nf0, Wv, bv, vb, ROWS, DD, DD, 0);

  // 4) relative-position MLP bias + speaker bias
  rel_spk_bias<<<ROWS, 128, 0, stream>>>(centers, spk, rW1, rb1, rW2, rb2, semb, biasNN);

  // 5) logits = qk^T/sqrt(D) + bias ; masked softmax (in place)
  qk_wmma<<<dim3(16, BATCH), 128, 0, stream>>>(qb, kb, biasNN, logit, rsqrtf((float)DD));
  softmax_mask<<<ROWS, 128, 0, stream>>>(logit, pmask);

  // 6) agg = (w @ v) * mask  -> reuse q buffer
  agg_wmma<<<dim3(64, BATCH), 128, 0, stream>>>(logit, vb, pmask, qb);

  // 7) o = agg @ Wo + bo -> reuse k buffer ; nf1 = LN(nf0 + o) in place
  gemm_big(qb, Wo, bo, kb, ROWS, DD, DD, 0);
  residual_ln<<<ROWS, 256, 0, stream>>>(nf0, kb, n1g, n1b, nullptr, nf0);

  // 8) FFN: h = relu(nf1@fW1+fb1); ff = h@fW2+fb2 -> reuse q buffer
  gemm_big(nf0, fW1, fb1, hb, ROWS, 2 * DD, DD, 1);
  gemm_big(hb,  fW2, fb2, qb, ROWS, DD, 2 * DD, 0);

  // 9) out = LN(nf1 + ff) * mask
  residual_ln<<<ROWS, 256, 0, stream>>>(nf0, qb, n2g, n2b, pmask, (float*)d_out);
}